// GATDeformerBlock_69569880261283
// MI455X (gfx1250) — compile-verified
//
#include <hip/hip_runtime.h>
#include <math.h>

typedef float v2f __attribute__((ext_vector_type(2)));
typedef float v8f __attribute__((ext_vector_type(8)));

static constexpr int NN = 5000;   // nodes
static constexpr int MP = 5008;   // rows padded to multiple of 16
static constexpr int EE = 40000;  // raw edges
static constexpr int EP = 45000;  // edges + self loops
static constexpr int NH = 8;      // heads

__device__ __forceinline__ float selu_f(float x) {
  const float scale = 1.0507009873554805f;
  const float alph  = 1.6732632423543772f;
  return x > 0.f ? scale * x : scale * alph * (expf(x) - 1.f);
}
__device__ __forceinline__ float lrelu_f(float x) { return x > 0.f ? x : 0.2f * x; }

// ---------------------------------------------------------------------------
// CSR build (graph is identical for all layers)
// ---------------------------------------------------------------------------
__global__ void build_edges(const int* __restrict__ eidx, int* __restrict__ src_all,
                            int* __restrict__ dst_all, int* __restrict__ deg) {
  int e = blockIdx.x * blockDim.x + threadIdx.x;
  if (e >= EP) return;
  int s, d;
  if (e < EE) { s = eidx[e]; d = eidx[EE + e]; }
  else        { s = e - EE;  d = e - EE; }          // self loops
  src_all[e] = s; dst_all[e] = d;
  atomicAdd(&deg[d], 1);
}

__global__ void scan_deg(const int* __restrict__ deg, int* __restrict__ rowptr,
                         int* __restrict__ cursor) {
  if (blockIdx.x == 0 && threadIdx.x == 0) {
    int acc = 0;
    for (int n = 0; n < NN; ++n) { rowptr[n] = acc; cursor[n] = acc; acc += deg[n]; }
    rowptr[NN] = acc;
  }
}

__global__ void fill_elist(const int* __restrict__ dst_all, int* __restrict__ cursor,
                           int* __restrict__ elist) {
  int e = blockIdx.x * blockDim.x + threadIdx.x;
  if (e >= EP) return;
  int pos = atomicAdd(&cursor[dst_all[e]], 1);
  elist[pos] = e;
}

// ---------------------------------------------------------------------------
// Pack weights into K-pair-interleaved layout, zero-padded to KP rows:
//   Bp[(k>>1)*NCOLS*2 + n*2 + (k&1)] = (k < K) ? W[k*NCOLS + n] : 0
// so a WMMA lane's two K-values are one contiguous 8-byte load.
// ---------------------------------------------------------------------------
__global__ void pack_weight(const float* __restrict__ W, float* __restrict__ Bp,
                            int K, int KP, int NCOLS) {
  int t = blockIdx.x * blockDim.x + threadIdx.x;
  if (t >= KP * NCOLS) return;
  int pair = t / (NCOLS * 2);
  int rem  = t - pair * (NCOLS * 2);
  int n    = rem >> 1;
  int k    = pair * 2 + (rem & 1);
  Bp[t] = (k < K) ? W[(size_t)k * NCOLS + n] : 0.f;
}

// ---------------------------------------------------------------------------
// Input MLP: af1[:, 0:2] = coord, af1[:, 2:258] = selu(data @ W0 + b0)
// ---------------------------------------------------------------------------
__global__ void init_af1(const float* __restrict__ data, const float* __restrict__ W0,
                         const float* __restrict__ b0, float* __restrict__ af1) {
  int t = blockIdx.x * blockDim.x + threadIdx.x;
  if (t >= NN * 256) return;
  int n = t >> 8, c = t & 255;
  float acc = b0[c];
#pragma unroll
  for (int k = 0; k < 10; ++k) acc += data[n * 10 + k] * W0[k * 256 + c];
  af1[(size_t)n * 260 + 2 + c] = selu_f(acc);
  if (c < 2) af1[(size_t)n * 260 + c] = data[n * 10 + c];
}

// ---------------------------------------------------------------------------
// GEMM via V_WMMA_F32_16X16X4_F32:  C[MP x NCOLS] = A[MP x KP] * B[KP x NCOLS]
// A zero-padded to KP (mult of 4) / MP (mult of 16); B packed by pack_weight.
// Block = 128 threads (4 waves) -> 16 x 256 C strip; each wave owns 4 tiles
// (16x64), reusing one A fragment per K-step against 4 b64-loaded B fragments.
// A tile staged in LDS (coalesced once per block, reused by all 4 waves).
// ---------------------------------------------------------------------------
template <int KP, int NCOLS>
__global__ __launch_bounds__(128) void gemm_wmma(const float* __restrict__ A,
                                                 const float* __restrict__ Bp,
                                                 float* __restrict__ C) {
  constexpr int LST = KP + 2;  // LDS row stride (mod 64 banks in {6,10}) -> conflict-free
  __shared__ float As[16 * LST];
  const int tid  = threadIdx.x;
  const int wave = tid >> 5;
  const int lane = tid & 31;
  const int lh   = lane >> 4;   // lane half selects K pair
  const int l    = lane & 15;
  const int m0   = blockIdx.x << 4;
  const int nbase = blockIdx.y * 256 + wave * 64;

  // cooperative, coalesced A-tile load into LDS
  for (int idx = tid; idx < 16 * KP; idx += 128) {
    int r = idx / KP, c = idx - r * KP;
    As[r * LST + c] = A[(size_t)(m0 + r) * KP + c];
  }
  __syncthreads();

  int  nc[4];
  bool vt[4];
#pragma unroll
  for (int i = 0; i < 4; ++i) {
    int n = nbase + i * 16;
    vt[i] = (n < NCOLS);
    nc[i] = vt[i] ? n : 0;  // clamp: loads stay in-bounds, EXEC stays all-ones
  }

  v8f acc[4] = {};
  const float* as_l = As + l * LST;
#pragma unroll 2
  for (int k = 0; k < KP; k += 4) {
    const int ka = k + (lh << 1);
    v2f a;
    a.x = as_l[ka];
    a.y = as_l[ka + 1];
    // packed B: pair row (ka>>1), column n+l -> 8 contiguous bytes
    const float* bprow = Bp + (size_t)(ka >> 1) * (NCOLS * 2) + l * 2;
#pragma unroll
    for (int i = 0; i < 4; ++i) {
      v2f b = *(const v2f*)(bprow + nc[i] * 2);
      acc[i] = __builtin_amdgcn_wmma_f32_16x16x4_f32(false, a, false, b, (short)0,
                                                     acc[i], false, false);
    }
  }

  // C/D layout: VGPR r, lane l(+16) -> element (M = r + 8*lh, N = l)
#pragma unroll
  for (int i = 0; i < 4; ++i) {
    if (!vt[i]) continue;  // wave-uniform predicate
    float* crow = C + (size_t)(m0 + (lh << 3)) * NCOLS + nc[i] + l;
#pragma unroll
    for (int r = 0; r < 8; ++r) crow[(size_t)r * NCOLS] = acc[i][r];
  }
}

// ---------------------------------------------------------------------------
// e_s[n,h] = sum_c h[n,h,c]*a_s[h,c]  (one wave per (n,head), shuffle reduce)
// ---------------------------------------------------------------------------
template <int DOUT>
__global__ void esed_kernel(const float* __restrict__ hbuf, const float* __restrict__ a_s,
                            const float* __restrict__ a_d,
                            float* __restrict__ es, float* __restrict__ ed) {
  int wid  = (blockIdx.x * blockDim.x + threadIdx.x) >> 5;
  int lane = threadIdx.x & 31;
  if (wid >= NN * NH) return;
  int n = wid >> 3, h = wid & 7;
  const float* hp  = hbuf + (size_t)n * (NH * DOUT) + h * DOUT;
  const float* asp = a_s + h * DOUT;
  const float* adp = a_d + h * DOUT;
  float s = 0.f, d = 0.f;
  for (int c = lane; c < DOUT; c += 32) {
    float hv = hp[c];
    s += hv * asp[c];
    d += hv * adp[c];
  }
#pragma unroll
  for (int off = 16; off > 0; off >>= 1) {
    s += __shfl_down(s, off, 32);
    d += __shfl_down(d, off, 32);
  }
  if (lane == 0) { es[wid] = s; ed[wid] = d; }
}

// ---------------------------------------------------------------------------
// Softmax over incoming edges, per (dst node, head). alpha[e,h] has a unique
// owner thread (dst(e),h) -> no atomics needed.
// ---------------------------------------------------------------------------
__global__ void attn_kernel(const int* __restrict__ rowptr, const int* __restrict__ elist,
                            const int* __restrict__ src_all, const float* __restrict__ es,
                            const float* __restrict__ ed, float* __restrict__ alpha) {
  int t = blockIdx.x * blockDim.x + threadIdx.x;
  if (t >= NN * NH) return;
  int n = t >> 3, h = t & 7;
  int beg = rowptr[n], end = rowptr[n + 1];
  float edv = ed[n * NH + h];
  float m = -1e30f;
  for (int j = beg; j < end; ++j) {
    int e = elist[j];
    float x = lrelu_f(es[src_all[e] * NH + h] + edv);
    m = fmaxf(m, x);
  }
  float sum = 0.f;
  for (int j = beg; j < end; ++j) {
    int e = elist[j];
    float x  = lrelu_f(es[src_all[e] * NH + h] + edv);
    float ex = expf(x - m);
    alpha[(size_t)e * NH + h] = ex;
    sum += ex;
  }
  float inv = 1.f / sum;
  for (int j = beg; j < end; ++j) alpha[(size_t)elist[j] * NH + h] *= inv;
}

// ---------------------------------------------------------------------------
// Feature aggregation: f[n,c] = selu( (1/H) sum_h sum_e alpha[e,h]*h[src,h,c] + bias[c] )
// written straight into the next layer's padded af at column offset col_off.
// ---------------------------------------------------------------------------
template <int DOUT>
__global__ void agg_feat(const int* __restrict__ rowptr, const int* __restrict__ elist,
                         const int* __restrict__ src_all, const float* __restrict__ alpha,
                         const float* __restrict__ hbuf, const float* __restrict__ bias,
                         float* __restrict__ af_next, int afw, int col_off) {
  int t = blockIdx.x * blockDim.x + threadIdx.x;
  if (t >= NN * DOUT) return;
  int n = t / DOUT, c = t - n * DOUT;
  int beg = rowptr[n], end = rowptr[n + 1];
  float acc = 0.f;
  for (int j = beg; j < end; ++j) {
    int e = elist[j];
    int s = src_all[e];
    const float* hp = hbuf + (size_t)s * (NH * DOUT) + c;
    const float* ap = alpha + (size_t)e * NH;
#pragma unroll
    for (int h = 0; h < NH; ++h) acc += ap[h] * hp[h * DOUT];
  }
  acc = acc * 0.125f + bias[c];
  af_next[(size_t)n * afw + col_off + c] = selu_f(acc);
}

// ---------------------------------------------------------------------------
// Coordinate update: oc = 0.2/H * sum_e (sum_h alpha[e,h]) * af_cur[src,0:2]
// then boundary conditions on af_cur[n,0:2].
// ---------------------------------------------------------------------------
__global__ void coord_agg(const int* __restrict__ rowptr, const int* __restrict__ elist,
                          const int* __restrict__ src_all, const float* __restrict__ alpha,
                          const float* __restrict__ af_cur, int afw,
                          float* __restrict__ cout) {
  int n = blockIdx.x * blockDim.x + threadIdx.x;
  if (n >= NN) return;
  int beg = rowptr[n], end = rowptr[n + 1];
  float acc0 = 0.f, acc1 = 0.f;
  for (int j = beg; j < end; ++j) {
    int e = elist[j];
    int s = src_all[e];
    const float* ap = alpha + (size_t)e * NH;
    float asum = 0.f;
#pragma unroll
    for (int h = 0; h < NH; ++h) asum += ap[h];
    acc0 += asum * af_cur[(size_t)s * afw + 0];
    acc1 += asum * af_cur[(size_t)s * afw + 1];
  }
  float oc0 = acc0 * 0.025f;  // 0.2 / 8
  float oc1 = acc1 * 0.025f;
  float a0 = af_cur[(size_t)n * afw + 0];
  float a1 = af_cur[(size_t)n * afw + 1];
  float c0 = (a0 == 0.f) ? 0.f : ((a0 == 1.f) ? 1.f : oc0);   // down, then up
  float c1 = (a1 == 1.f) ? 1.f : ((a1 == 0.f) ? 0.f : oc1);   // right, then left
  cout[n * 2 + 0] = c0;
  cout[n * 2 + 1] = c1;
}

// af_next[:,0:2] = c_new ; af_next[:,2:2+ncopy] = af_cur[:,0:ncopy]
__global__ void build_coords(const float* __restrict__ c_new, const float* __restrict__ af_cur,
                             int afw_cur, int ncopy, float* __restrict__ af_next, int afw_next) {
  int n = blockIdx.x * blockDim.x + threadIdx.x;
  if (n >= NN) return;
  af_next[(size_t)n * afw_next + 0] = c_new[n * 2 + 0];
  af_next[(size_t)n * afw_next + 1] = c_new[n * 2 + 1];
  for (int c = 0; c < ncopy; ++c)
    af_next[(size_t)n * afw_next + 2 + c] = af_cur[(size_t)n * afw_cur + c];
}

// ---------------------------------------------------------------------------
static inline int cdiv(long long a, long long b) { return (int)((a + b - 1) / b); }

extern "C" void kernel_launch(void* const* d_in, const int* in_sizes, int n_in,
                              void* d_out, int out_size, void* d_ws, size_t ws_size,
                              hipStream_t stream) {
  const float* data = (const float*)d_in[0];
  const int*   eidx = (const int*)d_in[1];
  const float* W0 = (const float*)d_in[2];
  const float* b0 = (const float*)d_in[3];
  const float* W1 = (const float*)d_in[4];
  const float* as1 = (const float*)d_in[5];
  const float* ad1 = (const float*)d_in[6];
  const float* bi1 = (const float*)d_in[7];
  const float* W2 = (const float*)d_in[8];
  const float* as2 = (const float*)d_in[9];
  const float* ad2 = (const float*)d_in[10];
  const float* bi2 = (const float*)d_in[11];
  const float* W3 = (const float*)d_in[12];
  const float* as3 = (const float*)d_in[13];
  const float* ad3 = (const float*)d_in[14];
  const float* bi3 = (const float*)d_in[15];
  const float* W4 = (const float*)d_in[16];
  const float* as4 = (const float*)d_in[17];
  const float* ad4 = (const float*)d_in[18];
  (void)n_in; (void)in_sizes; (void)ws_size; (void)out_size;
  float* out = (float*)d_out;

  // workspace carve-up (256B aligned blocks)
  char* p = (char*)d_ws;
  auto take = [&](size_t bytes) -> char* {
    char* r = p;
    p += (bytes + 255) & ~(size_t)255;
    return r;
  };
  int*   src_all = (int*)take((size_t)EP * 4);
  int*   dst_all = (int*)take((size_t)EP * 4);
  int*   deg     = (int*)take((size_t)NN * 4);
  int*   rowptr  = (int*)take((size_t)(NN + 1) * 4);
  int*   cursor  = (int*)take((size_t)NN * 4);
  int*   elist   = (int*)take((size_t)EP * 4);
  float* af1  = (float*)take((size_t)MP * 260 * 4);
  float* af2  = (float*)take((size_t)MP * 516 * 4);
  float* af3  = (float*)take((size_t)MP * 264 * 4);
  float* af4  = (float*)take((size_t)MP * 136 * 4);
  float* Bp1  = (float*)take((size_t)260 * 4096 * 4);
  float* Bp2  = (float*)take((size_t)516 * 2048 * 4);
  float* Bp3  = (float*)take((size_t)264 * 1024 * 4);
  float* Bp4  = (float*)take((size_t)136 * 160 * 4);
  float* hbuf = (float*)take((size_t)MP * 4096 * 4);
  float* es    = (float*)take((size_t)NN * NH * 4);
  float* ed    = (float*)take((size_t)NN * NH * 4);
  float* alpha = (float*)take((size_t)EP * NH * 4);
  float* c1 = (float*)take((size_t)NN * 2 * 4);
  float* c2 = (float*)take((size_t)NN * 2 * 4);
  float* c3 = (float*)take((size_t)NN * 2 * 4);

  // zero-init (graph-capturable memsets): padded rows/cols of af must be 0
  hipMemsetAsync(deg, 0, (size_t)NN * 4, stream);
  hipMemsetAsync(af1, 0, (size_t)MP * 260 * 4, stream);
  hipMemsetAsync(af2, 0, (size_t)MP * 516 * 4, stream);
  hipMemsetAsync(af3, 0, (size_t)MP * 264 * 4, stream);
  hipMemsetAsync(af4, 0, (size_t)MP * 136 * 4, stream);

  // CSR build (shared by all layers) + packed/padded weights
  build_edges<<<cdiv(EP, 256), 256, 0, stream>>>(eidx, src_all, dst_all, deg);
  scan_deg<<<1, 1, 0, stream>>>(deg, rowptr, cursor);
  fill_elist<<<cdiv(EP, 256), 256, 0, stream>>>(dst_all, cursor, elist);
  pack_weight<<<cdiv(260 * 4096, 256), 256, 0, stream>>>(W1, Bp1, 258, 260, 4096);
  pack_weight<<<cdiv(516 * 2048, 256), 256, 0, stream>>>(W2, Bp2, 516, 516, 2048);
  pack_weight<<<cdiv(264 * 1024, 256), 256, 0, stream>>>(W3, Bp3, 262, 264, 1024);
  pack_weight<<<cdiv(136 * 160, 256), 256, 0, stream>>>(W4, Bp4, 136, 136, 160);

  // input MLP
  init_af1<<<cdiv((long long)NN * 256, 256), 256, 0, stream>>>(data, W0, b0, af1);

  // ---- layer 1: din 258 (Kp 260), dout 512, Ncols 4096 ----
  gemm_wmma<260, 4096><<<dim3(MP / 16, 16), 128, 0, stream>>>(af1, Bp1, hbuf);
  esed_kernel<512><<<NN * NH / 8, 256, 0, stream>>>(hbuf, as1, ad1, es, ed);
  attn_kernel<<<cdiv(NN * NH, 256), 256, 0, stream>>>(rowptr, elist, src_all, es, ed, alpha);
  agg_feat<512><<<cdiv((long long)NN * 512, 256), 256, 0, stream>>>(rowptr, elist, src_all,
                                                                    alpha, hbuf, bi1, af2, 516, 4);
  coord_agg<<<cdiv(NN, 256), 256, 0, stream>>>(rowptr, elist, src_all, alpha, af1, 260, c1);
  build_coords<<<cdiv(NN, 256), 256, 0, stream>>>(c1, af1, 260, 2, af2, 516);

  // ---- layer 2: din 516, dout 256, Ncols 2048 ----
  gemm_wmma<516, 2048><<<dim3(MP / 16, 8), 128, 0, stream>>>(af2, Bp2, hbuf);
  esed_kernel<256><<<NN * NH / 8, 256, 0, stream>>>(hbuf, as2, ad2, es, ed);
  attn_kernel<<<cdiv(NN * NH, 256), 256, 0, stream>>>(rowptr, elist, src_all, es, ed, alpha);
  agg_feat<256><<<cdiv((long long)NN * 256, 256), 256, 0, stream>>>(rowptr, elist, src_all,
                                                                    alpha, hbuf, bi2, af3, 264, 6);
  coord_agg<<<cdiv(NN, 256), 256, 0, stream>>>(rowptr, elist, src_all, alpha, af2, 516, c2);
  build_coords<<<cdiv(NN, 256), 256, 0, stream>>>(c2, af2, 516, 4, af3, 264);

  // ---- layer 3: din 262 (Kp 264), dout 128, Ncols 1024 ----
  gemm_wmma<264, 1024><<<dim3(MP / 16, 4), 128, 0, stream>>>(af3, Bp3, hbuf);
  esed_kernel<128><<<NN * NH / 8, 256, 0, stream>>>(hbuf, as3, ad3, es, ed);
  attn_kernel<<<cdiv(NN * NH, 256), 256, 0, stream>>>(rowptr, elist, src_all, es, ed, alpha);
  agg_feat<128><<<cdiv((long long)NN * 128, 256), 256, 0, stream>>>(rowptr, elist, src_all,
                                                                    alpha, hbuf, bi3, af4, 136, 8);
  coord_agg<<<cdiv(NN, 256), 256, 0, stream>>>(rowptr, elist, src_all, alpha, af3, 264, c3);
  build_coords<<<cdiv(NN, 256), 256, 0, stream>>>(c3, af3, 264, 6, af4, 136);

  // ---- layer 4: din 136, dout 20, Ncols 160 (coords only; f4 unused) ----
  gemm_wmma<136, 160><<<dim3(MP / 16, 1), 128, 0, stream>>>(af4, Bp4, hbuf);
  esed_kernel<20><<<NN * NH / 8, 256, 0, stream>>>(hbuf, as4, ad4, es, ed);
  attn_kernel<<<cdiv(NN * NH, 256), 256, 0, stream>>>(rowptr, elist, src_all, es, ed, alpha);
  coord_agg<<<cdiv(NN, 256), 256, 0, stream>>>(rowptr, elist, src_all, alpha, af4, 136, out);
}